// DetBenchPredict_16441134809697
// MI455X (gfx1250) — compile-verified
//
#include <hip/hip_runtime.h>
#include <stdint.h>
#include <stddef.h>

#define KTOP   5000
#define MAXDET 100
#define NBINS  8192
#define BNDCAP 16384
#define NEGV   (-1e10f)

// Order-preserving bit key: descending float order == descending uint order.
static __device__ __forceinline__ uint32_t fkey(float f) {
  uint32_t u = __float_as_uint(f);
  return (u & 0x80000000u) ? ~u : (u | 0x80000000u);
}
static __device__ __forceinline__ float keyf(uint32_t k) {
  uint32_t u = (k & 0x80000000u) ? (k & 0x7fffffffu) : ~k;
  return __uint_as_float(u);
}

__global__ void initK(uint32_t* __restrict__ p, int n) {
  int i = blockIdx.x * blockDim.x + threadIdx.x;
  int st = gridDim.x * blockDim.x;
  for (; i < n; i += st) p[i] = 0u;
}

// info layout (uint32): [0..7]=threshold bin, [8..15]=count strictly above,
//                       [16..23]=cand counter, [24..31]=boundary counter

// Pass 1: per-image 8192-bin histogram of class logits.
// Double-buffered CDNA5 async global->LDS staging (ASYNCcnt) + LDS atomics.
__global__ void __launch_bounds__(256) histK(const float* __restrict__ src, int n4,
                                             uint32_t* __restrict__ ghist) {
  __shared__ uint32_t hist[NBINS];
  __shared__ float4 stage[8 * 4 * 32];  // [wave][buf(2)*slot(2)][lane] 16B each
  const int b = blockIdx.y;
  for (int i = threadIdx.x; i < NBINS; i += blockDim.x) hist[i] = 0u;
  __syncthreads();

  const uint64_t base = (uint64_t)(uintptr_t)src + (uint64_t)b * ((uint64_t)n4 * 16u);
  const uint32_t tid  = blockIdx.x * blockDim.x + threadIdx.x;
  const uint32_t nthr = gridDim.x * blockDim.x;
  const uint32_t lane = threadIdx.x & 31u;
  const uint32_t wv   = threadIdx.x >> 5;
  // Low 32 bits of a flat shared pointer are the LDS byte offset (ISA 10.2).
  const uint32_t lds0 = (uint32_t)(uintptr_t)&stage[(wv * 4u) * 32u + lane];

  // Full batches: batch m covers idx = tid + (2m+s)*nthr, s in {0,1};
  // in-range for every lane iff (2m+2)*nthr <= n4.
  const uint32_t NF = (uint32_t)n4 / (2u * nthr);

  auto issue = [&](uint32_t buf, uint32_t m) {
#pragma unroll
    for (uint32_t s = 0; s < 2u; ++s) {
      uint32_t idx = tid + (2u * m + s) * nthr;
      uint32_t la = lds0 + (buf * 2u + s) * 512u;
      uint32_t voff = idx * 16u;
      asm volatile("global_load_async_to_lds_b128 %0, %1, %2"
                   :: "v"(la), "v"(voff), "s"(base) : "memory");
    }
  };

  if (NF > 0) issue(0u, 0u);
  for (uint32_t m = 0; m < NF; ++m) {
    const uint32_t cur = m & 1u;
    if (m + 1u < NF) {
      issue(cur ^ 1u, m + 1u);
      // async loads complete in order: cnt<=2 => previous buffer landed in LDS
      asm volatile("s_wait_asynccnt 2" ::: "memory");
    } else {
      asm volatile("s_wait_asynccnt 0" ::: "memory");
    }
#pragma unroll
    for (uint32_t s = 0; s < 2u; ++s) {
      float4 v = stage[(wv * 4u + cur * 2u + s) * 32u + lane];
      atomicAdd(&hist[fkey(v.x) >> 19], 1u);
      atomicAdd(&hist[fkey(v.y) >> 19], 1u);
      atomicAdd(&hist[fkey(v.z) >> 19], 1u);
      atomicAdd(&hist[fkey(v.w) >> 19], 1u);
    }
  }
  // Tail (non-uniform remainder): plain vectorized loads.
  const float4* s4 = (const float4*)(src + (size_t)b * ((size_t)n4 * 4));
  for (uint32_t idx = tid + 2u * NF * nthr; idx < (uint32_t)n4; idx += nthr) {
    float4 v = s4[idx];
    atomicAdd(&hist[fkey(v.x) >> 19], 1u);
    atomicAdd(&hist[fkey(v.y) >> 19], 1u);
    atomicAdd(&hist[fkey(v.z) >> 19], 1u);
    atomicAdd(&hist[fkey(v.w) >> 19], 1u);
  }
  __syncthreads();
  uint32_t* gh = ghist + (size_t)b * NBINS;
  for (int i = threadIdx.x; i < NBINS; i += blockDim.x) {
    uint32_t c = hist[i];
    if (c) atomicAdd(&gh[i], c);
  }
}

// Descending-bin scan: find bin t with cum(bins>t) < K <= cum(bins>=t).
__global__ void __launch_bounds__(256) scanK(const uint32_t* __restrict__ ghist,
                                             uint32_t* __restrict__ info) {
  __shared__ uint32_t sh[NBINS];
  __shared__ uint32_t cs[256];
  const int b = blockIdx.x;
  const uint32_t* gh = ghist + (size_t)b * NBINS;
  for (int d = threadIdx.x; d < NBINS; d += 256) sh[d] = gh[NBINS - 1 - d];
  __syncthreads();
  uint32_t s = 0;
  const int base = threadIdx.x * 32;
  for (int j = 0; j < 32; ++j) s += sh[base + j];
  cs[threadIdx.x] = s;
  __syncthreads();
  if (threadIdx.x == 0) {
    uint32_t acc = 0;
    for (int i = 0; i < 256; ++i) { uint32_t x = cs[i]; cs[i] = acc; acc += x; }
  }
  __syncthreads();
  uint32_t cum = cs[threadIdx.x];
  for (int j = 0; j < 32; ++j) {
    uint32_t c = sh[base + j];
    if (cum < (uint32_t)KTOP && cum + c >= (uint32_t)KTOP) {
      info[b]     = (uint32_t)(NBINS - 1 - (base + j));  // threshold bin
      info[8 + b] = cum;                                  // strictly-above count
    }
    cum += c;
  }
}

// Pass 2: collect above-threshold candidates + boundary-bin elements.
// fidx = flattened index into cls_all.reshape(B,-1) after NCHW->NHWC reorder.
template <int CELLSZ>
__global__ void __launch_bounds__(256) collectK(
    const float* __restrict__ src, int n4, int lvloff, uint32_t* __restrict__ info,
    int* __restrict__ cand_idx, float* __restrict__ cand_val,
    uint32_t* __restrict__ bnd_key, int* __restrict__ bnd_idx) {
  const int b = blockIdx.y;
  const uint32_t tb = info[b];
  const float4* s4 = (const float4*)(src + (size_t)b * ((size_t)n4 * 4));
  uint32_t tid = blockIdx.x * blockDim.x + threadIdx.x;
  uint32_t nthr = gridDim.x * blockDim.x;
  for (uint32_t i = tid; i < (uint32_t)n4; i += nthr) {
    float4 v = s4[i];
    uint32_t p0 = i * 4u;                        // NCHW flat = ch*CELLSZ + cell
    uint32_t ch = p0 / (uint32_t)CELLSZ;         // CELLSZ % 4 == 0, same ch for 4
    uint32_t cell0 = p0 - ch * (uint32_t)CELLSZ;
    uint32_t k = ch / 90u;
    uint32_t c = ch - k * 90u;
    uint32_t abase = (uint32_t)lvloff + cell0 * 9u + k;
    float va[4] = {v.x, v.y, v.z, v.w};
#pragma unroll
    for (uint32_t j = 0; j < 4u; ++j) {
      uint32_t key = fkey(va[j]);
      uint32_t bin = key >> 19;
      if (bin >= tb) {
        uint32_t fidx = (abase + j * 9u) * 90u + c;
        if (bin > tb) {
          uint32_t sl = atomicAdd(&info[16 + b], 1u);
          cand_idx[(size_t)b * KTOP + sl] = (int)fidx;
          cand_val[(size_t)b * KTOP + sl] = va[j];
        } else {
          uint32_t sl = atomicAdd(&info[24 + b], 1u);
          if (sl < BNDCAP) {
            bnd_key[(size_t)b * BNDCAP + sl] = key;
            bnd_idx[(size_t)b * BNDCAP + sl] = (int)fidx;
          }
        }
      }
    }
  }
}

// Exact rank selection inside the boundary bin; tie order (key desc, idx asc)
// matches jax.lax.top_k's stable tie-breaking.
__global__ void __launch_bounds__(1024) resolveK(
    const uint32_t* __restrict__ info, const uint32_t* __restrict__ bnd_key,
    const int* __restrict__ bnd_idx, int* __restrict__ cand_idx,
    float* __restrict__ cand_val) {
  __shared__ uint32_t tk[1024];
  __shared__ int ti[1024];
  const int b = blockIdx.x;
  uint32_t n = info[24 + b]; if (n > BNDCAP) n = BNDCAP;
  const uint32_t nab = info[8 + b];
  const uint32_t need = (uint32_t)KTOP - nab;
  uint32_t myk[16]; int myi[16]; uint32_t rk[16];
  int mc = 0;
  for (uint32_t i = threadIdx.x; i < n && mc < 16; i += 1024) {
    myk[mc] = bnd_key[(size_t)b * BNDCAP + i];
    myi[mc] = bnd_idx[(size_t)b * BNDCAP + i];
    rk[mc] = 0; ++mc;
  }
  for (uint32_t c0 = 0; c0 < n; c0 += 1024) {
    uint32_t mch = n - c0; if (mch > 1024u) mch = 1024u;
    __syncthreads();
    if (threadIdx.x < mch) {
      tk[threadIdx.x] = bnd_key[(size_t)b * BNDCAP + c0 + threadIdx.x];
      ti[threadIdx.x] = bnd_idx[(size_t)b * BNDCAP + c0 + threadIdx.x];
    }
    __syncthreads();
    for (int e = 0; e < mc; ++e) {
      uint32_t ke = myk[e]; int ie = myi[e]; uint32_t r = 0;
      for (uint32_t j = 0; j < mch; ++j) {
        uint32_t kj = tk[j];
        if (kj > ke || (kj == ke && ti[j] < ie)) ++r;
      }
      rk[e] += r;
    }
  }
  for (int e = 0; e < mc; ++e) {
    if (rk[e] < need) {
      uint32_t slot = nab + rk[e];   // ranks are distinct -> unique slots
      cand_idx[(size_t)b * KTOP + slot] = myi[e];
      cand_val[(size_t)b * KTOP + slot] = keyf(myk[e]);
    }
  }
}

// Box gather + decode + sigmoid.
__global__ void __launch_bounds__(256) decodeK(
    const int* __restrict__ cand_idx, const float* __restrict__ cand_val,
    const float* __restrict__ anch,
    const float* __restrict__ b0, const float* __restrict__ b1,
    const float* __restrict__ b2, const float* __restrict__ b3,
    const float* __restrict__ b4,
    float* __restrict__ boxes, float* __restrict__ scores, int* __restrict__ clsA) {
  int i = blockIdx.x * blockDim.x + threadIdx.x;
  if (i >= 8 * KTOP) return;
  int b = i / KTOP;
  int fidx = cand_idx[i];
  float logit = cand_val[i];
  int a = fidx / 90;
  int c = fidx - a * 90;
  const float* bp; int W; int rem;
  if (a < 82944)       { bp = b0; W = 96; rem = a; }
  else if (a < 103680) { bp = b1; W = 48; rem = a - 82944; }
  else if (a < 108864) { bp = b2; W = 24; rem = a - 103680; }
  else if (a < 110160) { bp = b3; W = 12; rem = a - 108864; }
  else                 { bp = b4; W = 6;  rem = a - 110160; }
  int cell = rem / 9;
  int k = rem - cell * 9;
  int y = cell / W;
  int x = cell - y * W;
  size_t cs = (size_t)W * W;
  const float* pb = bp + ((size_t)b * 36 + (size_t)(k * 4)) * cs + (size_t)y * W + x;
  float ty = pb[0];
  float tx = pb[cs];
  float th = pb[2 * cs];
  float tw = pb[3 * cs];
  float4 an = *(const float4*)(anch + (size_t)a * 4);  // yxyx
  float yca = (an.x + an.z) * 0.5f;
  float xca = (an.y + an.w) * 0.5f;
  float ha = an.z - an.x;
  float wa = an.w - an.y;
  float w = expf(tw) * wa;
  float h = expf(th) * ha;
  float yc = ty * ha + yca;
  float xc = tx * wa + xca;
  float* ob = boxes + (size_t)i * 4;
  ob[0] = xc - w * 0.5f;
  ob[1] = yc - h * 0.5f;
  ob[2] = xc + w * 0.5f;
  ob[3] = yc + h * 0.5f;
  scores[i] = 1.0f / (1.0f + expf(-logit));
  clsA[i] = c;
}

// Argmax comparator reproducing jnp.argmax over top-k-ordered arrays:
// higher sigmoid first, then higher logit, then smaller flattened index.
static __device__ __forceinline__ bool better(float av, float al, int af,
                                              float bv, float bl, int bf) {
  if (av != bv) return av > bv;
  if (al != bl) return al > bl;
  return af < bf;
}

__global__ void __launch_bounds__(1024) nmsK(
    const float* __restrict__ boxes, const float* __restrict__ scores,
    const int* __restrict__ clsA, const int* __restrict__ cand_idx,
    const float* __restrict__ cand_val, float* __restrict__ bbg,
    float* __restrict__ out) {
  __shared__ float sv[1024];
  __shared__ float sl[1024];
  __shared__ int   sf[1024];
  __shared__ int   si[1024];
  __shared__ float live[KTOP];
  __shared__ float ar[KTOP];
  const int b = blockIdx.x;
  const int t = threadIdx.x;
  const float* B = boxes + (size_t)b * KTOP * 4;
  float* BB = bbg + (size_t)b * KTOP * 4;

  // max over decoded boxes -> class offset scale
  float m = -3.402823466e38f;
  for (int i = t; i < KTOP * 4; i += 1024) m = fmaxf(m, B[i]);
  sv[t] = m;
  __syncthreads();
  for (int s = 512; s > 0; s >>= 1) {
    if (t < s) sv[t] = fmaxf(sv[t], sv[t + s]);
    __syncthreads();
  }
  const float offscale = sv[0] + 1.0f;
  __syncthreads();

  for (int i = t; i < KTOP; i += 1024) {
    float off = (float)clsA[b * KTOP + i] * offscale;
    float x1 = B[i * 4 + 0] + off;
    float y1 = B[i * 4 + 1] + off;
    float x2 = B[i * 4 + 2] + off;
    float y2 = B[i * 4 + 3] + off;
    BB[i * 4 + 0] = x1; BB[i * 4 + 1] = y1;
    BB[i * 4 + 2] = x2; BB[i * 4 + 3] = y2;
    ar[i] = (x2 - x1) * (y2 - y1);
    live[i] = scores[b * KTOP + i];
  }
  __syncthreads();

  for (int it = 0; it < MAXDET; ++it) {
    float bv = -3.402823466e38f, bl = -3.402823466e38f;
    int bf = 0x7fffffff, bi = 0;
    for (int i = t; i < KTOP; i += 1024) {
      float v = live[i];
      float lg = cand_val[b * KTOP + i];
      int fd = cand_idx[b * KTOP + i];
      if (better(v, lg, fd, bv, bl, bf)) { bv = v; bl = lg; bf = fd; bi = i; }
    }
    sv[t] = bv; sl[t] = bl; sf[t] = bf; si[t] = bi;
    __syncthreads();
    for (int s = 512; s > 0; s >>= 1) {
      if (t < s && better(sv[t + s], sl[t + s], sf[t + s], sv[t], sl[t], sf[t])) {
        sv[t] = sv[t + s]; sl[t] = sl[t + s]; sf[t] = sf[t + s]; si[t] = si[t + s];
      }
      __syncthreads();
    }
    const int idx = si[0];
    const bool ok = sv[0] > 0.0f;
    __syncthreads();
    if (t == 0) {
      float* o = out + ((size_t)b * MAXDET + it) * 6;
      if (ok) {
        o[0] = B[idx * 4 + 0]; o[1] = B[idx * 4 + 1];
        o[2] = B[idx * 4 + 2]; o[3] = B[idx * 4 + 3];
        o[4] = scores[b * KTOP + idx];
        o[5] = (float)(clsA[b * KTOP + idx] + 1);
      } else {
        o[0] = o[1] = o[2] = o[3] = o[4] = o[5] = 0.0f;
      }
    }
    const float bx1 = BB[idx * 4 + 0], by1 = BB[idx * 4 + 1];
    const float bx2 = BB[idx * 4 + 2], by2 = BB[idx * 4 + 3];
    const float ai = ar[idx];
    for (int i = t; i < KTOP; i += 1024) {
      float ix = fminf(BB[i * 4 + 2], bx2) - fmaxf(BB[i * 4 + 0], bx1);
      ix = fmaxf(0.0f, ix);
      float iy = fminf(BB[i * 4 + 3], by2) - fmaxf(BB[i * 4 + 1], by1);
      iy = fmaxf(0.0f, iy);
      float inter = ix * iy;
      float iou = inter / (ar[i] + ai - inter + 1e-8f);
      if (iou > 0.5f) live[i] = NEGV;
    }
    if (t == 0) live[idx] = NEGV;
    __syncthreads();
  }
}

extern "C" void kernel_launch(void* const* d_in, const int* in_sizes, int n_in,
                              void* d_out, int out_size, void* d_ws, size_t ws_size,
                              hipStream_t stream) {
  (void)in_sizes; (void)n_in; (void)out_size;
  // setup_inputs order is interleaved: cls0, box0, cls1, box1, ..., anchors
  const float* cls[5] = {(const float*)d_in[0], (const float*)d_in[2],
                         (const float*)d_in[4], (const float*)d_in[6],
                         (const float*)d_in[8]};
  const float* box[5] = {(const float*)d_in[1], (const float*)d_in[3],
                         (const float*)d_in[5], (const float*)d_in[7],
                         (const float*)d_in[9]};
  const float* anch = (const float*)d_in[10];
  float* out = (float*)d_out;

  uint8_t* ws = (uint8_t*)d_ws;
  size_t off = 0;
  auto take = [&](size_t bytes) -> void* {
    void* p = ws + off;
    off += (bytes + 255) & ~(size_t)255;
    return p;
  };
  uint32_t* hist   = (uint32_t*)take((size_t)8 * NBINS * 4);
  uint32_t* info   = (uint32_t*)take(32 * 4);
  int*      cand_i = (int*)take((size_t)8 * KTOP * 4);
  float*    cand_v = (float*)take((size_t)8 * KTOP * 4);
  uint32_t* bnd_k  = (uint32_t*)take((size_t)8 * BNDCAP * 4);
  int*      bnd_i  = (int*)take((size_t)8 * BNDCAP * 4);
  float*    boxesW = (float*)take((size_t)8 * KTOP * 4 * 4);
  float*    scoresW= (float*)take((size_t)8 * KTOP * 4);
  int*      clsW   = (int*)take((size_t)8 * KTOP * 4);
  float*    bbW    = (float*)take((size_t)8 * KTOP * 4 * 4);
  if (off > ws_size) return;

  static const int HW[5] = {96, 48, 24, 12, 6};
  static const int LVLOFF[5] = {0, 82944, 103680, 108864, 110160};

  initK<<<128, 256, 0, stream>>>(hist, 8 * NBINS);
  initK<<<1, 64, 0, stream>>>(info, 32);

  for (int l = 0; l < 5; ++l) {
    int n4 = 810 * HW[l] * HW[l] / 4;
    int bx = (n4 + 2047) / 2048; if (bx > 512) bx = 512; if (bx < 1) bx = 1;
    histK<<<dim3(bx, 8), 256, 0, stream>>>(cls[l], n4, hist);
  }
  scanK<<<8, 256, 0, stream>>>(hist, info);
  for (int l = 0; l < 5; ++l) {
    int n4 = 810 * HW[l] * HW[l] / 4;
    int bx = (n4 + 2047) / 2048; if (bx > 512) bx = 512; if (bx < 1) bx = 1;
    dim3 g(bx, 8);
    switch (l) {
      case 0: collectK<9216><<<g, 256, 0, stream>>>(cls[0], n4, LVLOFF[0], info, cand_i, cand_v, bnd_k, bnd_i); break;
      case 1: collectK<2304><<<g, 256, 0, stream>>>(cls[1], n4, LVLOFF[1], info, cand_i, cand_v, bnd_k, bnd_i); break;
      case 2: collectK<576 ><<<g, 256, 0, stream>>>(cls[2], n4, LVLOFF[2], info, cand_i, cand_v, bnd_k, bnd_i); break;
      case 3: collectK<144 ><<<g, 256, 0, stream>>>(cls[3], n4, LVLOFF[3], info, cand_i, cand_v, bnd_k, bnd_i); break;
      case 4: collectK<36  ><<<g, 256, 0, stream>>>(cls[4], n4, LVLOFF[4], info, cand_i, cand_v, bnd_k, bnd_i); break;
    }
  }
  resolveK<<<8, 1024, 0, stream>>>(info, bnd_k, bnd_i, cand_i, cand_v);
  decodeK<<<(8 * KTOP + 255) / 256, 256, 0, stream>>>(cand_i, cand_v, anch,
      box[0], box[1], box[2], box[3], box[4], boxesW, scoresW, clsW);
  nmsK<<<8, 1024, 0, stream>>>(boxesW, scoresW, clsW, cand_i, cand_v, bbW, out);
}